// DKD_76304388981021
// MI455X (gfx1250) — compile-verified
//
#include <hip/hip_runtime.h>
#include <hip/hip_bf16.h>

typedef __attribute__((ext_vector_type(2))) float v2f;
typedef __attribute__((ext_vector_type(8))) float v8f;

#define IMG_H 1024
#define IMG_W 1024
#define IMG_HW (IMG_H * IMG_W)
#define NTILES 65536   // 256*256 4x4 tiles
#define TOPK 4096
#define CAPG 4096      // capacity for strictly-greater candidates (power of 2)
#define CAPE 1024      // capacity for equal-to-threshold candidates (power of 2)
#define BORDER 3       // RADIUS + 1

// Output layout (floats): keypoints [0,8192), descriptors [8192,270336), scores [270336,274432)
#define OUT_DESC_OFF 8192
#define OUT_SCORE_OFF 270336

// ---------------------------------------------------------------------------
// Kernel 1: 4x4 tile max + argmax with border zeroing. One thread per tile.
// Coalesced: adjacent threads read adjacent float4 in the same image row.
// ---------------------------------------------------------------------------
__global__ __launch_bounds__(256) void dkd_tiles(const float* __restrict__ scores,
                                                 float* __restrict__ vals,
                                                 unsigned* __restrict__ pos) {
    int t = blockIdx.x * 256 + threadIdx.x;     // tile id, 0..65535
    int tr = t >> 8;                            // tile row
    int tc = t & 255;                           // tile col
    float best = -1.0f;
    int bidx = 0;
#pragma unroll
    for (int ky = 0; ky < 4; ++ky) {
        int y = tr * 4 + ky;
        float4 v4 = *(const float4*)(scores + (size_t)y * IMG_W + tc * 4);
        float e[4] = {v4.x, v4.y, v4.z, v4.w};
        bool rowbad = (y < BORDER) || (y >= IMG_H - BORDER);
#pragma unroll
        for (int kx = 0; kx < 4; ++kx) {
            int x = tc * 4 + kx;
            bool bad = rowbad || (x < BORDER) || (x >= IMG_W - BORDER);
            float v = bad ? 0.0f : e[kx];
            if (v > best) { best = v; bidx = ky * 4 + kx; }  // strict >: first index wins
        }
    }
    vals[t] = best;
    int row = tr * 4 + (bidx >> 2);
    int col = tc * 4 + (bidx & 3);
    pos[t] = ((unsigned)row << 16) | (unsigned)col;
}

// ---------------------------------------------------------------------------
// Kernel 2: exact stable top-4096 of 65536 values. Single workgroup (1024 thr).
// Non-negative floats -> uint bit pattern is order-preserving.
// 3-pass radix select (11/11/10 bits) for exact threshold, then bitonic sorts.
// ---------------------------------------------------------------------------
__global__ __launch_bounds__(1024) void dkd_topk(const float* __restrict__ vals,
                                                 const unsigned* __restrict__ pos,
                                                 float* __restrict__ sel_val,
                                                 unsigned* __restrict__ sel_pos) {
    __shared__ unsigned hist[2048];
    __shared__ unsigned long long candG[CAPG];  // (key<<16) | (65535-idx)
    __shared__ unsigned candE[CAPE];            // idx of key == T
    __shared__ unsigned sh_prefix, sh_rem, sh_cntG, sh_cntE;

    int tid = threadIdx.x;
    if (tid == 0) { sh_prefix = 0u; sh_rem = TOPK; sh_cntG = 0u; sh_cntE = 0u; }
    __syncthreads();

    const int shifts[3]  = {21, 10, 0};
    const int bits[3]    = {11, 11, 10};
    for (int p = 0; p < 3; ++p) {
        int shift = shifts[p];
        int nb = 1 << bits[p];
        unsigned bmask = (unsigned)(nb - 1);
        int top = shift + bits[p];
        unsigned pmask = (top >= 32) ? 0u : (0xFFFFFFFFu << top);
        for (int b = tid; b < nb; b += 1024) hist[b] = 0u;
        __syncthreads();
        unsigned prefix = sh_prefix;
        for (int i = tid; i < NTILES; i += 1024) {
            unsigned key = __float_as_uint(vals[i]);
            if ((key & pmask) == prefix)
                atomicAdd(&hist[(key >> shift) & bmask], 1u);
        }
        __syncthreads();
        if (tid == 0) {
            unsigned rem = sh_rem, cum = 0u;
            int chosen = 0;
            for (int b = nb - 1; b >= 0; --b) {
                unsigned c = hist[b];
                if (cum + c >= rem) { chosen = b; break; }
                cum += c;
            }
            sh_prefix = prefix | ((unsigned)chosen << shift);
            sh_rem = rem - cum;
        }
        __syncthreads();
    }

    unsigned T = sh_prefix;  // exact 32-bit threshold key

    // init candidate arrays
    for (int i = tid; i < CAPG; i += 1024) candG[i] = 0ull;
    if (tid < CAPE) candE[tid] = 0xFFFFFFFFu;
    __syncthreads();

    // collect
    for (int i = tid; i < NTILES; i += 1024) {
        unsigned key = __float_as_uint(vals[i]);
        if (key > T) {
            unsigned s = atomicAdd(&sh_cntG, 1u);
            if (s < CAPG)
                candG[s] = ((unsigned long long)key << 16) | (unsigned long long)(65535u - (unsigned)i);
        } else if (key == T) {
            unsigned s = atomicAdd(&sh_cntE, 1u);
            if (s < CAPE) candE[s] = (unsigned)i;
        }
    }
    __syncthreads();
    unsigned G = sh_cntG;
    if (G > TOPK) G = TOPK;  // safety clamp

    // bitonic sort candG DESCENDING by composite (key desc, idx asc). Pads (0) sink.
    for (int k = 2; k <= CAPG; k <<= 1) {
        for (int j = k >> 1; j > 0; j >>= 1) {
            for (int i = tid; i < CAPG; i += 1024) {
                int l = i ^ j;
                if (l > i) {
                    unsigned long long a = candG[i], b = candG[l];
                    bool seg = ((i & k) == 0);
                    if ((seg && a < b) || (!seg && a > b)) { candG[i] = b; candG[l] = a; }
                }
            }
            __syncthreads();
        }
    }

    // bitonic sort candE ASCENDING by index. Pads (0xFFFFFFFF) sink.
    for (int k = 2; k <= CAPE; k <<= 1) {
        for (int j = k >> 1; j > 0; j >>= 1) {
            int i = tid;
            if (i < CAPE) {
                int l = i ^ j;
                if (l > i) {
                    unsigned a = candE[i], b = candE[l];
                    bool seg = ((i & k) == 0);
                    if ((seg && a > b) || (!seg && a < b)) { candE[i] = b; candE[l] = a; }
                }
            }
            __syncthreads();
        }
    }

    float Tval = __uint_as_float(T);
    for (int kk = tid; kk < TOPK; kk += 1024) {
        unsigned idx;
        float v;
        if ((unsigned)kk < G) {
            unsigned long long comp = candG[kk];
            idx = 65535u - (unsigned)(comp & 0xFFFFull);
            v = __uint_as_float((unsigned)(comp >> 16));
        } else {
            idx = candE[kk - G];
            if (idx > 65535u) idx = 0u;  // safety
            v = Tval;
        }
        sel_val[kk] = v;
        sel_pos[kk] = pos[idx];
    }
}

// ---------------------------------------------------------------------------
// Kernel 3: gather + WMMA-based L2 normalization + outputs.
// 4 waves/block, 16 keypoints per wave. Wave32 A-layout of V_WMMA_F32_16X16X4_F32:
//   lanes 0-15:  M = lane,    VGPR0=K0, VGPR1=K1
//   lanes 16-31: M = lane-16, VGPR0=K2, VGPR1=K3
// which equals the B-layout of its own transpose, so acc = wmma(a, a, acc)
// accumulates the Gram matrix D*D^T; the diagonal is sum-of-squares.
// ---------------------------------------------------------------------------
__global__ __launch_bounds__(128) void dkd_gather(const float* __restrict__ dmap,
                                                  const float* __restrict__ sel_val,
                                                  const unsigned* __restrict__ sel_pos,
                                                  float* __restrict__ out) {
    __shared__ float sdesc[4][16 * 64];  // per-wave 16 keypoints x 64 channels
    __shared__ float snorm[4][16];

    int tid = threadIdx.x;
    int wv = tid >> 5;
    int lane = tid & 31;
    int kp_local = lane & 15;
    int half = lane >> 4;                         // 0: K0/K1, 1: K2/K3
    int kp = blockIdx.x * 64 + wv * 16 + kp_local;

    unsigned p = sel_pos[kp];
    int row = (int)(p >> 16);
    int col = (int)(p & 0xFFFFu);
    const float* dbase = dmap + (size_t)row * IMG_W + col;

    float* sd = &sdesc[wv][kp_local * 64];
    v8f acc = {0.f, 0.f, 0.f, 0.f, 0.f, 0.f, 0.f, 0.f};

#pragma unroll
    for (int k = 0; k < 16; ++k) {               // 16 K-chunks of 4 channels
        int ch = k * 4 + half * 2;
        float v0 = dbase[(size_t)ch * IMG_HW];
        float v1 = dbase[(size_t)(ch + 1) * IMG_HW];
        sd[ch] = v0;
        sd[ch + 1] = v1;
        v2f a = {v0, v1};
        acc = __builtin_amdgcn_wmma_f32_16x16x4_f32(
            /*neg_a=*/false, a, /*neg_b=*/false, a,
            /*c_mod=*/(short)0, acc, /*reuse_a=*/false, /*reuse_b=*/false);
    }

    // Diagonal of the 16x16 Gram matrix:
    //   M=r   (r<8)  -> lane r,      VGPR r
    //   M=r+8 (r<8)  -> lane r+24,   VGPR r
#pragma unroll
    for (int r = 0; r < 8; ++r) {
        if (lane == r)      snorm[wv][r]     = acc[r];
        if (lane == r + 24) snorm[wv][r + 8] = acc[r];
    }
    __syncthreads();

    float rinv = __frsqrt_rn(snorm[wv][kp_local]);

    // Each lane writes half (32 channels) of its keypoint's normalized descriptor.
    float* od = out + OUT_DESC_OFF + (size_t)kp * 64 + half * 32;
    const float* sdh = sd + half * 32;
#pragma unroll
    for (int j = 0; j < 32; j += 4) {
        float4 v = *(const float4*)(sdh + j);
        v.x *= rinv; v.y *= rinv; v.z *= rinv; v.w *= rinv;
        *(float4*)(od + j) = v;
    }

    if (half == 0) {
        out[kp * 2 + 0] = (float)col / 1023.0f * 2.0f - 1.0f;
        out[kp * 2 + 1] = (float)row / 1023.0f * 2.0f - 1.0f;
        out[OUT_SCORE_OFF + kp] = sel_val[kp];
    }
}

// ---------------------------------------------------------------------------
extern "C" void kernel_launch(void* const* d_in, const int* in_sizes, int n_in,
                              void* d_out, int out_size, void* d_ws, size_t ws_size,
                              hipStream_t stream) {
    const float* scores = (const float*)d_in[0];   // (1,1,1024,1024) f32
    const float* dmap   = (const float*)d_in[1];   // (1,64,1024,1024) f32
    float* out = (float*)d_out;

    char* ws = (char*)d_ws;
    float*    vals    = (float*)(ws);                 // 65536 f32
    unsigned* pos     = (unsigned*)(ws + 262144);     // 65536 u32
    float*    sel_val = (float*)(ws + 524288);        // 4096 f32
    unsigned* sel_pos = (unsigned*)(ws + 540672);     // 4096 u32

    dkd_tiles<<<256, 256, 0, stream>>>(scores, vals, pos);
    dkd_topk<<<1, 1024, 0, stream>>>(vals, pos, sel_val, sel_pos);
    dkd_gather<<<64, 128, 0, stream>>>(dmap, sel_val, sel_pos, out);
}